// TokenEmbedding_55259049230486
// MI455X (gfx1250) — compile-verified
//
#include <hip/hip_runtime.h>
#include <hip/hip_bf16.h>

// Token embedding gather: out[i, d] = weight[d, tokens[i]] + bias[d]
//   tokens: [16384] int32, weight: [1024, 50257] fp32 (row-major), bias: [1024] fp32
//   out:    [16384, 1024] fp32
//
// Memory-bound gather. Weight-column reads are 4B @ 201KB stride (one 128B line
// per element per token), but each line serves ~10 tokens and the 206MB table
// ~fits the 192MB L2, so HBM read ~= table size (~206MB) + 64MB NT output
// writes => ~270MB @ 23.3TB/s ~= 12us floor. WMMA is inapplicable (a one-hot
// matmul would inflate FLOPs by ~VOCAB); this kernel optimizes the cache
// policy and scalar/vector instruction mix instead.

#define VOCAB    50257
#define NUM_DIM  1024

typedef float v4f __attribute__((ext_vector_type(4)));

constexpr int TOK_PER_BLOCK = 4;
constexpr int THREADS       = 256;   // 8 wave32 per block; 256*4 = 1024 dims

__device__ __forceinline__ void emit_token(int i, int tok, int d0,
                                           const float* __restrict__ wrow,
                                           const v4f& b,
                                           float* __restrict__ out)
{
    // 4 gather loads share one address; offsets k*VOCAB*4 < 2^23 fold into
    // the 24-bit instruction offset.
    const float* __restrict__ p = wrow + tok;
    v4f r;
    r.x = p[0 * (size_t)VOCAB] + b.x;
    r.y = p[1 * (size_t)VOCAB] + b.y;
    r.z = p[2 * (size_t)VOCAB] + b.z;
    r.w = p[3 * (size_t)VOCAB] + b.w;
    // Non-temporal b128 store: write-once output must not evict the table
    // from the 192MB L2.
    v4f* dst = (v4f*)(out + (size_t)i * NUM_DIM + d0);
    __builtin_nontemporal_store(r, dst);
}

__global__ __launch_bounds__(THREADS)
void token_embed_gather(const int*   __restrict__ tokens,
                        const float* __restrict__ weight,
                        const float* __restrict__ bias,
                        float*       __restrict__ out,
                        int ntok)
{
    // CDNA5 marker (free: TENSORcnt is already 0).
    __builtin_amdgcn_s_wait_tensorcnt(0);

    const int base = blockIdx.x * TOK_PER_BLOCK;
    // Warm the NEXT block's token line (this block's tokens are s_loaded now).
    __builtin_prefetch(tokens + base + TOK_PER_BLOCK, 0, 0); // global_prefetch_b8

    const int d0 = threadIdx.x << 2;           // this thread's 4 dims: d0..d0+3

    // Bias: coalesced b128 load, reused across all tokens of this block.
    const v4f b = *(const v4f*)(bias + d0);

    // Base of this thread's 4 weight rows.
    const float* __restrict__ wrow = weight + (size_t)d0 * VOCAB;

    if (base + TOK_PER_BLOCK <= ntok) {
        // Full block (the only path for ntok % 4 == 0): one uniform 16B
        // scalar fetch of all 4 token ids, no per-token guards.
        const int4 tk = *(const int4*)(tokens + base);   // uniform -> s_load
        emit_token(base + 0, tk.x, d0, wrow, b, out);
        emit_token(base + 1, tk.y, d0, wrow, b, out);
        emit_token(base + 2, tk.z, d0, wrow, b, out);
        emit_token(base + 3, tk.w, d0, wrow, b, out);
    } else {
        // Tail block (defensive; unused when ntok is a multiple of 4).
        for (int t = 0; t < TOK_PER_BLOCK; ++t) {
            const int i = base + t;
            if (i < ntok)
                emit_token(i, tokens[i], d0, wrow, b, out);
        }
    }
}

extern "C" void kernel_launch(void* const* d_in, const int* in_sizes, int n_in,
                              void* d_out, int out_size, void* d_ws, size_t ws_size,
                              hipStream_t stream)
{
    const int*   tokens = (const int*)  d_in[0];   // [B*S] int32
    const float* weight = (const float*)d_in[1];   // [NUM_DIM, VOCAB] fp32
    const float* bias   = (const float*)d_in[2];   // [NUM_DIM] fp32
    float*       out    = (float*)      d_out;     // [B*S, NUM_DIM] fp32

    const int ntok   = in_sizes[0];
    const int blocks = (ntok + TOK_PER_BLOCK - 1) / TOK_PER_BLOCK;

    token_embed_gather<<<dim3(blocks), dim3(THREADS), 0, stream>>>(
        tokens, weight, bias, out, ntok);
}